// Upsample_72619307041391
// MI455X (gfx1250) — compile-verified
//
#include <hip/hip_runtime.h>

typedef float v2f __attribute__((ext_vector_type(2)));
typedef float v8f __attribute__((ext_vector_type(8)));

#define IN_H  512
#define IN_W  512
#define OUT_H 1024
#define OUT_W 1024

// Upsample-x2 (nearest) + 1x16 horizontal FIR (SAME, pad_lo=7), cast as
// out(16x16) = U(16x32) x Toeplitz(32x16) via 8 chained V_WMMA_F32_16X16X4_F32.
__global__ __launch_bounds__(256) void upsample_blur_wmma(
    const float* __restrict__ x, const float* __restrict__ kern,
    float* __restrict__ out)
{
    __shared__ float s_in[8][80];   // 8 input rows x 80 cols (zero-filled borders)
    __shared__ float s_w[16];       // flattened 4x4 blur kernel -> 16 FIR taps

    const int tid  = threadIdx.x;
    const int lane = tid & 31;
    const int wv   = tid >> 5;

    const int xbase = blockIdx.x * 128;   // output col base for workgroup
    const int y0    = blockIdx.y * 16;    // output row base (even)
    const int n     = blockIdx.z;

    const int iy0   = y0 >> 1;            // 8 input rows feed 16 output rows
    const int icol0 = (xbase >> 1) - 4;   // leftmost staged input column

    if (tid < 16) s_w[tid] = kern[tid];

    // Stage 8 x 80 input columns into LDS, zero outside [0, IN_W).
    const float* xrow0 = x + ((size_t)n * IN_H + iy0) * IN_W;
    for (int idx = tid; idx < 8 * 80; idx += 256) {
        int r  = idx / 80;
        int c  = idx - r * 80;
        int gc = icol0 + c;
        float v = 0.0f;
        if (gc >= 0 && gc < IN_W) v = xrow0[(size_t)r * IN_W + gc];
        s_in[r][c] = v;
    }
    __syncthreads();

    const int half = lane >> 4;       // which half-wave (K offset / M offset)
    const int mrow = lane & 15;       // M row for A, N column for B
    const int koff = half * 2;

    // Per-lane Toeplitz B operands: T[k][c] = w[k-c] for 0 <= k-c < 16, else 0.
    v2f b[8];
    #pragma unroll
    for (int m = 0; m < 8; ++m) {
        int k0 = 4 * m + koff;
        int t0 = k0 + 0 - mrow;
        int t1 = k0 + 1 - mrow;
        b[m][0] = (t0 >= 0 && t0 < 16) ? s_w[t0] : 0.0f;
        b[m][1] = (t1 >= 0 && t1 < 16) ? s_w[t1] : 0.0f;
    }

    const int cbase = xbase + wv * 16;    // this wave's 16 output columns
    const int srow  = mrow >> 1;          // staged input row for output row mrow
    v8f c = {};

    #pragma unroll
    for (int m = 0; m < 8; ++m) {
        int k0  = 4 * m + koff;
        int ux0 = cbase - 7 + k0;         // upsampled-domain x coordinate
        v2f a;
        a[0] = s_in[srow][(ux0 >> 1) - icol0];         // up[x] = in[x>>1]
        a[1] = s_in[srow][((ux0 + 1) >> 1) - icol0];
        c = __builtin_amdgcn_wmma_f32_16x16x4_f32(
                /*neg_a=*/false, a, /*neg_b=*/false, b[m],
                /*c_mod=*/(short)0, c, /*reuse_a=*/false, /*reuse_b=*/false);
    }

    // D layout: VGPR v holds row (half*8 + v), column mrow.
    float* orow = out + ((size_t)n * OUT_H + (y0 + half * 8)) * OUT_W
                      + cbase + mrow;
    #pragma unroll
    for (int v = 0; v < 8; ++v) {
        orow[(size_t)v * OUT_W] = c[v];
    }
}

extern "C" void kernel_launch(void* const* d_in, const int* in_sizes, int n_in,
                              void* d_out, int out_size, void* d_ws, size_t ws_size,
                              hipStream_t stream) {
    const float* x = (const float*)d_in[0];   // (N, 512, 512, 1) fp32
    const float* k = (const float*)d_in[1];   // 4x4 kernel -> 16 taps
    float* out = (float*)d_out;               // (N, 1024, 1024, 1) fp32

    int N = in_sizes[0] / (IN_H * IN_W);      // 32
    dim3 grid(OUT_W / 128, OUT_H / 16, N);
    upsample_blur_wmma<<<grid, dim3(256), 0, stream>>>(x, k, out);
}